// GraphConvolutionLayer_481036337832
// MI455X (gfx1250) — compile-verified
//
#include <hip/hip_runtime.h>
#include <hip/hip_bf16.h>

// ---------------------------------------------------------------------------
// GCN layer: out = relu( spmm(adj, features @ weight) )
//   features: [50000, 512] f32   weight: [512, 256] f32
//   edges: COO (vals f32, rows i32, cols i32), E = 1,600,000
// Pipeline: zero(out) -> WMMA f32 GEMM (support in d_ws) -> edge-parallel
//           atomic scatter SpMM -> ReLU.
// ---------------------------------------------------------------------------

#define N_NODES 50000
#define IN_F    512
#define OUT_F   256

typedef float v2f __attribute__((ext_vector_type(2)));
typedef float v8f __attribute__((ext_vector_type(8)));

// K chunk and LDS strides (dwords). Stride 36: rows 16B-aligned (144 B) and
// WMMA fragment reads are bank-conflict-free (36*16 == 9*64 -> 16 distinct
// banks across lanes 0-15; lane halves differ mod 4 -> no cross-half hit).
#define KC        32
#define LF_STRIDE 36            // feature tile row stride  (32 + 4)
#define LW_STRIDE 36            // transposed weight column stride (32 + 4)

// ---------------------------------------------------------------------------
// Kernel 1: support = features @ weight via V_WMMA_F32_16X16X4_F32.
// Block = 256 threads = 8 waves. Block tile: 64 rows x 256 cols.
// Wave (mb, nh): mb = wv>>1 picks a 16-row band, nh = wv&1 picks a 128-col
// half -> 8 accumulators of 16x16 (8 x v8f = 64 VGPRs).
// Weight tile is stored TRANSPOSED in LDS so each B fragment is one aligned
// ds_load_b64; two k-steps are processed per iteration with ALL fragments
// hoisted before the 16-WMMA burst (one dscnt wait per 16 WMMAs).
// ---------------------------------------------------------------------------
__global__ __launch_bounds__(256) void gcn_gemm_wmma(
    const float* __restrict__ feat,   // [N_NODES, IN_F]
    const float* __restrict__ w,      // [IN_F, OUT_F]
    float* __restrict__ support)      // [N_NODES, OUT_F]
{
    __shared__ float lwT[OUT_F * LW_STRIDE];  // 36,864 B  (lwT[n][k])
    __shared__ float lf [64    * LF_STRIDE];  //  9,216 B  (lf[r][k])

    const int tid  = threadIdx.x;
    const int lane = tid & 31;             // wave32
    const int wv   = tid >> 5;             // 0..7
    const int mb   = wv >> 1;              // 0..3 : 16-row band
    const int nh   = wv & 1;               // 0..1 : 128-col half
    const int m    = lane & 15;
    const int kh   = lane >> 4;            // 0..1 : lane half
    const int row0 = blockIdx.x * 64;

    v8f acc[8] = {};

    for (int kb = 0; kb < IN_F; kb += KC) {
        // --- weight chunk, transposed into LDS: lwT[n][k] = w[kb+k][n] ---
        // Thread tid owns column n = tid; reads 4 k-consecutive values per
        // group (lane-contiguous in n -> coalesced 128B lines, L2-resident),
        // stores one float4 along k (16B aligned: (36n + 4i)*4 % 16 == 0).
        #pragma unroll
        for (int i = 0; i < KC / 4; ++i) {
            float4 v;
            v.x = w[(kb + 4 * i + 0) * OUT_F + tid];
            v.y = w[(kb + 4 * i + 1) * OUT_F + tid];
            v.z = w[(kb + 4 * i + 2) * OUT_F + tid];
            v.w = w[(kb + 4 * i + 3) * OUT_F + tid];
            *(float4*)(&lwT[tid * LW_STRIDE + 4 * i]) = v;
        }
        // --- feature tile [64 x KC], coalesced float4, 2 per thread ---
        #pragma unroll
        for (int i = 0; i < 2; ++i) {
            int id = i * 256 + tid;            // 0..511 float4s
            int r  = id >> 3;                  // 0..63
            int c4 = id & 7;                   // 0..7
            int g  = row0 + r;
            if (g > N_NODES - 1) g = N_NODES - 1;   // clamp (stores masked)
            float4 v = *(const float4*)(feat + (long long)g * IN_F + kb + c4 * 4);
            *(float4*)(&lf[r * LF_STRIDE + c4 * 4]) = v;
        }
        __syncthreads();

        // --- 4 double-K-steps; each: hoist 1 A-pair + 8 B-pairs (b64 LDS
        //     loads, batchable into ds_load_2addr_b64), then 16 WMMAs. ---
        #pragma unroll
        for (int k2 = 0; k2 < KC; k2 += 8) {
            const float* aBase = &lf[(mb * 16 + m) * LF_STRIDE + k2 + 2 * kh];
            v2f a0 = *(const v2f*)(aBase);
            v2f a1 = *(const v2f*)(aBase + 4);
            v2f b0[8], b1[8];
            #pragma unroll
            for (int t = 0; t < 8; ++t) {
                const float* bBase =
                    &lwT[(nh * 128 + t * 16 + m) * LW_STRIDE + k2 + 2 * kh];
                b0[t] = *(const v2f*)(bBase);
                b1[t] = *(const v2f*)(bBase + 4);
            }
            #pragma unroll
            for (int t = 0; t < 8; ++t) {
                acc[t] = __builtin_amdgcn_wmma_f32_16x16x4_f32(
                    false, a0, false, b0[t], (short)0, acc[t], false, false);
                acc[t] = __builtin_amdgcn_wmma_f32_16x16x4_f32(
                    false, a1, false, b1[t], (short)0, acc[t], false, false);
            }
        }
        __syncthreads();
    }

    // --- store D: VGPR r -> rows r (lanes 0-15) / r+8 (lanes 16-31) ---
    #pragma unroll
    for (int t = 0; t < 8; ++t) {
        int n = nh * 128 + t * 16 + m;
        #pragma unroll
        for (int r = 0; r < 8; ++r) {
            int row = row0 + mb * 16 + r + 8 * kh;
            if (row < N_NODES)
                support[(long long)row * OUT_F + n] = acc[t][r];
        }
    }
}

// ---------------------------------------------------------------------------
// Kernel 2: edge-parallel SpMM scatter.
// 64 threads (2 waves) per edge; each thread handles one float4 (4 cols).
// support row gather is contiguous 512B per half-warp (L2-resident: 51 MB
// fits the 192 MB L2). Scatter via native f32 atomic add (no-return).
// ---------------------------------------------------------------------------
__global__ __launch_bounds__(256) void gcn_spmm_scatter(
    const float* __restrict__ support,
    const float* __restrict__ evals,
    const int*   __restrict__ erows,
    const int*   __restrict__ ecols,
    float* __restrict__ out,
    int nEdges)
{
    long long idx = (long long)blockIdx.x * blockDim.x + threadIdx.x;
    int e  = (int)(idx >> 6);          // edge id
    int f4 = (int)(idx & 63);          // float4 slot within the 256 cols
    if (e >= nEdges) return;

    int   r = erows[e];
    int   c = ecols[e];
    float v = evals[e];

    float4 s = *(const float4*)(support + (long long)c * OUT_F + f4 * 4);
    float* dst = out + (long long)r * OUT_F + f4 * 4;
    unsafeAtomicAdd(dst + 0, v * s.x);
    unsafeAtomicAdd(dst + 1, v * s.y);
    unsafeAtomicAdd(dst + 2, v * s.z);
    unsafeAtomicAdd(dst + 3, v * s.w);
}

// ---------------------------------------------------------------------------
// Kernel 3/4: zero-init and ReLU epilogue (float4 streaming).
// ---------------------------------------------------------------------------
__global__ __launch_bounds__(256) void gcn_zero(float4* __restrict__ out, int n4)
{
    int i = blockIdx.x * blockDim.x + threadIdx.x;
    if (i < n4) out[i] = make_float4(0.f, 0.f, 0.f, 0.f);
}

__global__ __launch_bounds__(256) void gcn_relu(float4* __restrict__ out, int n4)
{
    int i = blockIdx.x * blockDim.x + threadIdx.x;
    if (i < n4) {
        float4 v = out[i];
        v.x = fmaxf(v.x, 0.f);
        v.y = fmaxf(v.y, 0.f);
        v.z = fmaxf(v.z, 0.f);
        v.w = fmaxf(v.w, 0.f);
        out[i] = v;
    }
}

// ---------------------------------------------------------------------------
extern "C" void kernel_launch(void* const* d_in, const int* in_sizes, int n_in,
                              void* d_out, int out_size, void* d_ws, size_t ws_size,
                              hipStream_t stream)
{
    const float* feat  = (const float*)d_in[0];   // [50000, 512]
    const float* w     = (const float*)d_in[1];   // [512, 256]
    const float* evals = (const float*)d_in[2];   // [E]
    const int*   erows = (const int*)d_in[3];     // [E]
    const int*   ecols = (const int*)d_in[4];     // [E]
    float*       out   = (float*)d_out;           // [50000, 256]
    float*       support = (float*)d_ws;          // [50000, 256] = 51.2 MB

    const int nEdges = in_sizes[2];
    const int n4 = out_size / 4;                  // out_size = 12,800,000

    // 1) zero output (d_out is poisoned before timing)
    gcn_zero<<<(n4 + 255) / 256, 256, 0, stream>>>((float4*)out, n4);

    // 2) support = features @ weight  (f32 WMMA)
    const int gemmBlocks = (N_NODES + 63) / 64;   // 782
    gcn_gemm_wmma<<<gemmBlocks, 256, 0, stream>>>(feat, w, support);

    // 3) scatter-add over edges
    long long spThreads = (long long)nEdges * 64;
    int spBlocks = (int)((spThreads + 255) / 256);
    gcn_spmm_scatter<<<spBlocks, 256, 0, stream>>>(support, evals, erows, ecols,
                                                   out, nEdges);

    // 4) ReLU in place
    gcn_relu<<<(n4 + 255) / 256, 256, 0, stream>>>((float4*)out, n4);
}